// SIGReg_28913719836695
// MI455X (gfx1250) — compile-verified
//
#include <hip/hip_runtime.h>
#include <hip/hip_bf16.h>
#include <math.h>

typedef float v2f __attribute__((ext_vector_type(2)));
typedef float v8f __attribute__((ext_vector_type(8)));

#define B_N 65536
#define D_N 768
#define S_N 64
#define T_N 17
#define DT_STEP (2.0f / 17.0f)

// workspace layout (float offsets)
#define OFF_DT    0u                     // 64*768 transposed directions
#define OFF_PROJ  49152u                 // 65536*64 projection
#define OFF_ACC   (49152u + 4194304u)    // accumulators (zeroed per launch)
#define ACC_DIMSUM  0u                   // 768
#define ACC_DIMSQ   768u                 // 768
#define ACC_COLSUM  1536u                // 64
#define ACC_COLSQ   1600u                // 64
#define ACC_ECFC    1664u                // 17*64
#define ACC_ECFS    2752u                // 17*64
#define ACC_TOTAL   3840u

__device__ __forceinline__ void atomAddG(float* p, float v) {
  __hip_atomic_fetch_add(p, v, __ATOMIC_RELAXED, __HIP_MEMORY_SCOPE_AGENT);
}
__device__ __forceinline__ void atomAddL(float* p, float v) {
  __hip_atomic_fetch_add(p, v, __ATOMIC_RELAXED, __HIP_MEMORY_SCOPE_WORKGROUP);
}

// ---------------------------------------------------------------- init: zero accumulators
__global__ void k_init(float* __restrict__ acc) {
  for (unsigned i = threadIdx.x; i < ACC_TOTAL; i += blockDim.x) acc[i] = 0.0f;
}

// ---------------------------------------------------------------- transpose dirs [768][64] -> Dt [64][768]
__global__ void k_transpose(const float* __restrict__ dirs, float* __restrict__ Dt) {
  unsigned idx = blockIdx.x * 256u + threadIdx.x;   // 192 blocks * 256 = 49152
  if (idx < (unsigned)(S_N * D_N)) {
    unsigned s = idx / D_N, d = idx % D_N;
    Dt[idx] = dirs[d * S_N + s];
  }
}

// ---------------------------------------------------------------- per-dim sums over B (coalesced)
__global__ void k_dimstats(const float* __restrict__ z, float* __restrict__ acc) {
  unsigned dchunk = blockIdx.x % 3u;        // 3 * 256 dims
  unsigned rchunk = blockIdx.x / 3u;        // 64 row chunks of 1024
  unsigned d = dchunk * 256u + threadIdx.x;
  const float* p = z + (size_t)rchunk * 1024u * D_N + d;
  float s = 0.0f, q = 0.0f;
#pragma unroll 4
  for (int r = 0; r < 1024; ++r) {
    float x = p[(size_t)r * D_N];
    s += x;
    q = fmaf(x, x, q);
  }
  atomAddG(acc + ACC_DIMSUM + d, s);
  atomAddG(acc + ACC_DIMSQ + d, q);
}

// ---------------------------------------------------------------- GEMM: proj = z @ dirs via V_WMMA_F32_16X16X4_F32
// block = 256 threads (8 waves); wave handles 16 rows x all 64 cols; grid = 512 blocks
__global__ void __launch_bounds__(256) k_gemm(const float* __restrict__ A,
                                              const float* __restrict__ Bt,
                                              float* __restrict__ P) {
  const int wave = threadIdx.x >> 5;
  const int lane = threadIdx.x & 31;
  const int half = lane >> 4;    // 0: K=k0,k0+1  1: K=k0+2,k0+3 (ISA A/B fragment layout)
  const int lr   = lane & 15;
  const int m0   = blockIdx.x * 128 + wave * 16;

  const float* arow = A + (size_t)(m0 + lr) * D_N + half * 2;
  const float* brow = Bt + (size_t)lr * D_N + half * 2;   // + ct*16*768 per column tile

  v8f acc0 = {}, acc1 = {}, acc2 = {}, acc3 = {};

#pragma unroll 8
  for (int k0 = 0; k0 < D_N; k0 += 4) {
    if ((k0 & 31) == 0) __builtin_prefetch(arow + k0 + 128, 0, 1);  // global_prefetch_b8
    v2f a  = *(const v2f*)(arow + k0);
    v2f b0 = *(const v2f*)(brow + k0);
    v2f b1 = *(const v2f*)(brow + 16 * D_N + k0);
    v2f b2 = *(const v2f*)(brow + 32 * D_N + k0);
    v2f b3 = *(const v2f*)(brow + 48 * D_N + k0);
    acc0 = __builtin_amdgcn_wmma_f32_16x16x4_f32(false, a, false, b0, (short)0, acc0, false, false);
    acc1 = __builtin_amdgcn_wmma_f32_16x16x4_f32(false, a, false, b1, (short)0, acc1, false, false);
    acc2 = __builtin_amdgcn_wmma_f32_16x16x4_f32(false, a, false, b2, (short)0, acc2, false, false);
    acc3 = __builtin_amdgcn_wmma_f32_16x16x4_f32(false, a, false, b3, (short)0, acc3, false, false);
  }

  // C/D layout: VGPR j -> M = j (lanes 0-15) / M = 8+j (lanes 16-31), N = lr (+16*ct)
#pragma unroll
  for (int j = 0; j < 8; ++j) {
    int row = m0 + half * 8 + j;
    float* pr = P + (size_t)row * S_N + lr;
    pr[0]  = acc0[j];
    pr[16] = acc1[j];
    pr[32] = acc2[j];
    pr[48] = acc3[j];
  }
}

// ---------------------------------------------------------------- per-column sums of proj
__global__ void k_colstats(const float* __restrict__ proj, float* __restrict__ acc) {
  __shared__ float redS[256];
  __shared__ float redQ[256];
  const int col = threadIdx.x & 63;
  const int g   = threadIdx.x >> 6;       // 4 row groups
  const float* p = proj + (size_t)blockIdx.x * 1024u * S_N + (size_t)g * S_N + col;
  float s = 0.0f, q = 0.0f;
#pragma unroll 4
  for (int r = 0; r < 256; ++r) {
    float x = p[(size_t)r * 4u * S_N];
    s += x;
    q = fmaf(x, x, q);
  }
  redS[threadIdx.x] = s;
  redQ[threadIdx.x] = q;
  __syncthreads();
  if (g == 0) {
    float ss = redS[col] + redS[col + 64] + redS[col + 128] + redS[col + 192];
    float qq = redQ[col] + redQ[col + 64] + redQ[col + 128] + redQ[col + 192];
    atomAddG(acc + ACC_COLSUM + col, ss);
    atomAddG(acc + ACC_COLSQ + col, qq);
  }
}

// ---------------------------------------------------------------- ECF sums via Chebyshev recurrence
__global__ void __launch_bounds__(256) k_ecf(const float* __restrict__ proj,
                                             float* __restrict__ acc) {
  __shared__ float lacc[2 * T_N * S_N];   // [cos 17*64][sin 17*64]
  for (unsigned i = threadIdx.x; i < 2u * T_N * S_N; i += 256u) lacc[i] = 0.0f;

  const int col = threadIdx.x & 63;
  const int g   = threadIdx.x >> 6;

  const float csum = acc[ACC_COLSUM + col];
  const float csq  = acc[ACC_COLSQ + col];
  const float mean = csum * (1.0f / (float)B_N);
  float var = (csq - csum * csum * (1.0f / (float)B_N)) * (1.0f / (float)(B_N - 1));
  var = fmaxf(var, 0.0f);
  const float inv = 1.0f / (sqrtf(var) + 1e-8f);

  __syncthreads();

  float accC[T_N], accS[T_N];
#pragma unroll
  for (int k = 0; k < T_N; ++k) { accC[k] = 0.0f; accS[k] = 0.0f; }

  int row = blockIdx.x * 512 + g;         // 128 blocks * 512 rows
  for (int i = 0; i < 128; ++i) {
    float p  = proj[(size_t)row * S_N + col];
    float th = (p - mean) * inv * DT_STEP;
    float s1, c1;
    __sincosf(th, &s1, &c1);
    float tc = 2.0f * c1;
    float cm2 = 1.0f, sm2 = 0.0f;
    float cm1 = c1, sm1 = s1;
    accC[0] += c1; accS[0] += s1;
#pragma unroll
    for (int k = 1; k < T_N; ++k) {
      float ck = fmaf(tc, cm1, -cm2);
      float sk = fmaf(tc, sm1, -sm2);
      accC[k] += ck; accS[k] += sk;
      cm2 = cm1; sm2 = sm1; cm1 = ck; sm1 = sk;
    }
    row += 4;
  }

#pragma unroll
  for (int k = 0; k < T_N; ++k) {
    atomAddL(&lacc[k * S_N + col], accC[k]);
    atomAddL(&lacc[T_N * S_N + k * S_N + col], accS[k]);
  }
  __syncthreads();
  for (unsigned i = threadIdx.x; i < 2u * T_N * S_N; i += 256u)
    atomAddG(acc + ACC_ECFC + i, lacc[i]);   // ECFC and ECFS are contiguous
}

// ---------------------------------------------------------------- final scalar assembly
__global__ void k_final(const float* __restrict__ acc, float* __restrict__ out) {
  __shared__ float red[256];
  const unsigned t = threadIdx.x;

  float vf = 0.0f;
  for (unsigned d = t; d < D_N; d += 256u) {
    float s1 = acc[ACC_DIMSUM + d];
    float var = (acc[ACC_DIMSQ + d] - s1 * s1 * (1.0f / (float)B_N)) * (1.0f / (float)(B_N - 1));
    var = fmaxf(var, 0.0f);
    vf += fmaxf(1.0f - sqrtf(var), 0.0f);
  }

  float tot = 0.0f;
  for (unsigned i = t; i < (unsigned)(T_N * S_N); i += 256u) {
    unsigned k = i >> 6;                         // 0..16 -> t = (k+1)*dt
    float tk  = (float)(k + 1) * DT_STEP;
    float er  = acc[ACC_ECFC + i] * (1.0f / (float)B_N);
    float ei  = acc[ACC_ECFS + i] * (1.0f / (float)B_N);
    float tcf = __expf(-0.5f * tk * tk);
    float val = er * er + ei * ei - 2.0f * er * tcf + tcf * tcf;
    float w   = DT_STEP * ((k == 0u || k == (unsigned)(T_N - 1)) ? 0.5f : 1.0f);
    tot += w * val;
  }

  red[t] = vf * (1.0f / (float)D_N) + tot * (1.0f / (float)S_N);
  __syncthreads();
  for (unsigned off = 128u; off > 0u; off >>= 1) {
    if (t < off) red[t] += red[t + off];
    __syncthreads();
  }
  if (t == 0) out[0] = red[0];
}

// ----------------------------------------------------------------
extern "C" void kernel_launch(void* const* d_in, const int* in_sizes, int n_in,
                              void* d_out, int out_size, void* d_ws, size_t ws_size,
                              hipStream_t stream) {
  (void)in_sizes; (void)n_in; (void)out_size; (void)ws_size;
  const float* emb  = (const float*)d_in[0];   // [65536,768] fp32
  const float* dirs = (const float*)d_in[1];   // [768,64] fp32
  float* ws   = (float*)d_ws;
  float* Dt   = ws + OFF_DT;
  float* proj = ws + OFF_PROJ;
  float* acc  = ws + OFF_ACC;
  float* out  = (float*)d_out;

  k_init<<<1, 256, 0, stream>>>(acc);
  k_transpose<<<192, 256, 0, stream>>>(dirs, Dt);
  k_gemm<<<512, 256, 0, stream>>>(emb, Dt, proj);
  k_dimstats<<<192, 256, 0, stream>>>(emb, acc);
  k_colstats<<<64, 256, 0, stream>>>(proj, acc);
  k_ecf<<<128, 256, 0, stream>>>(proj, acc);
  k_final<<<1, 256, 0, stream>>>(acc, out);
}